// MyModel_14259291423267
// MI455X (gfx1250) — compile-verified
//
#include <hip/hip_runtime.h>

// ARMA GNN (2 layers, T=1, K=1) for MI455X / gfx1250.
// Dense transforms: V_WMMA_F32_16X16X4_F32, one wave per 16-node tile (exact
// fp32 math; dense FLOPs are negligible so no precision downcast needed).
// Edge phase: one thread per edge, float4 gather + 16x global_atomic_add_f32.
// Cache policy: edge_index (51MB int64) and norm (12.8MB) are streamed with
// non-temporal loads; node feature buffers (6.4MB each) stay resident in the
// 192MB L2 so gathers/atomics never touch HBM.

typedef __attribute__((ext_vector_type(2))) float v2f;
typedef __attribute__((ext_vector_type(8))) float v8f;

// ---------------------------------------------------------------- utilities
__global__ void zero_f32_kernel(float* __restrict__ p, int n) {
  int i = blockIdx.x * blockDim.x + threadIdx.x;
  if (i < n) p[i] = 0.0f;
}

__global__ void deg_accum_kernel(const long long* __restrict__ col,
                                 float* __restrict__ deg, int E) {
  int e = blockIdx.x * blockDim.x + threadIdx.x;
  if (e < E) {
    int c = (int)__builtin_nontemporal_load(col + e);
    atomicAdd(&deg[c], 1.0f);
  }
}

__global__ void dinv_kernel(float* __restrict__ deg, int n) {
  int i = blockIdx.x * blockDim.x + threadIdx.x;
  if (i < n) {
    float d = deg[i];
    deg[i] = (d > 0.0f) ? rsqrtf(d) : 0.0f;  // in-place deg -> deg^-1/2
  }
}

__global__ void norm_kernel(const long long* __restrict__ row,
                            const long long* __restrict__ col,
                            const float* __restrict__ dinv,
                            float* __restrict__ norm, int E) {
  int e = blockIdx.x * blockDim.x + threadIdx.x;
  if (e < E) {
    int r = (int)__builtin_nontemporal_load(row + e);
    int c = (int)__builtin_nontemporal_load(col + e);
    __builtin_nontemporal_store(dinv[r] * dinv[c], norm + e);
  }
}

// -------------------------------------------------- dense transform (WMMA)
// One wave computes, for a tile of 16 nodes:
//   XW[16x16]  = X[16xFIN] @ W[FINx16]          (edge-message operand)
//   ACC[16x16] = X[16xFIN] @ V[FINx16] + b      (root term, scatter target)
// via V_WMMA_F32_16X16X4_F32 accumulated over FIN/4 K-steps.
template <int FIN>
__global__ void arma_dense_kernel(const float* __restrict__ X,
                                  const float* __restrict__ W,
                                  const float* __restrict__ V,
                                  const float* __restrict__ b,
                                  float* __restrict__ XW,
                                  float* __restrict__ ACC,
                                  int N) {
  const int lane = threadIdx.x & 31;
  const int wave = (int)((blockIdx.x * blockDim.x + threadIdx.x) >> 5);
  const int base = wave * 16;
  if (base >= N) return;            // wave-uniform: EXEC stays all-1s for WMMA

  const int h = lane >> 4;          // half-wave: selects K within a 4-wide slice
  const int n = lane & 15;          // output column (and A row index)
  int rowIdx = base + n;            // node whose row this lane supplies to A
  if (rowIdx > N - 1) rowIdx = N - 1;  // address clamp only; tail stores guarded

  const float bn = b[n];

  v8f cW = {};
  v8f cV = {};

#pragma unroll
  for (int s = 0; s < FIN / 4; ++s) {
    const int k = 4 * s + 2 * h;    // global K of this lane's first element
    // A 16x4 slice: lane holds X[m][k], X[m][k+1]
    v2f a;
    a.x = X[(size_t)rowIdx * FIN + k];
    a.y = X[(size_t)rowIdx * FIN + k + 1];
    // B 4x16 slice: lane holds W[k][n], W[k+1][n]
    v2f bw, bv;
    bw.x = W[k * 16 + n];
    bw.y = W[(k + 1) * 16 + n];
    bv.x = V[k * 16 + n];
    bv.y = V[(k + 1) * 16 + n];
    cW = __builtin_amdgcn_wmma_f32_16x16x4_f32(false, a, false, bw,
                                               (short)0, cW, false, false);
    cV = __builtin_amdgcn_wmma_f32_16x16x4_f32(false, a, false, bv,
                                               (short)0, cV, false, false);
  }

  // C/D layout: VGPR r -> row r (lanes 0-15) / row r+8 (lanes 16-31), col n.
  // Fast path (wave-uniform branch): full tile -> unconditional stores, no
  // per-store EXEC save/restore. Only the final partial tile takes the guard.
  if (base + 16 <= N) {
#pragma unroll
    for (int r = 0; r < 8; ++r) {
      const int outRow = base + r + 8 * h;
      XW[(size_t)outRow * 16 + n]  = cW[r];
      ACC[(size_t)outRow * 16 + n] = cV[r] + bn;
    }
  } else {
#pragma unroll
    for (int r = 0; r < 8; ++r) {
      const int outRow = base + r + 8 * h;
      if (outRow < N) {
        XW[(size_t)outRow * 16 + n]  = cW[r];
        ACC[(size_t)outRow * 16 + n] = cV[r] + bn;
      }
    }
  }
}

// ----------------------------------------------------- edge scatter (L2 atomics)
__global__ void scatter_kernel(const long long* __restrict__ row,
                               const long long* __restrict__ col,
                               const float* __restrict__ norm,
                               const float* __restrict__ src,
                               float* __restrict__ dst, int E) {
  int e = blockIdx.x * blockDim.x + threadIdx.x;
  if (e >= E) return;
  const int r   = (int)__builtin_nontemporal_load(row + e);
  const int c   = (int)__builtin_nontemporal_load(col + e);
  const float nm = __builtin_nontemporal_load(norm + e);
  // Gathered node rows are L2-resident (regular temporal policy).
  const float4* s4 = (const float4*)(src + (size_t)r * 16);
  float4 m0 = s4[0], m1 = s4[1], m2 = s4[2], m3 = s4[3];
  float* d = dst + (size_t)c * 16;
  atomicAdd(d + 0,  nm * m0.x);  atomicAdd(d + 1,  nm * m0.y);
  atomicAdd(d + 2,  nm * m0.z);  atomicAdd(d + 3,  nm * m0.w);
  atomicAdd(d + 4,  nm * m1.x);  atomicAdd(d + 5,  nm * m1.y);
  atomicAdd(d + 6,  nm * m1.z);  atomicAdd(d + 7,  nm * m1.w);
  atomicAdd(d + 8,  nm * m2.x);  atomicAdd(d + 9,  nm * m2.y);
  atomicAdd(d + 10, nm * m2.z);  atomicAdd(d + 11, nm * m2.w);
  atomicAdd(d + 12, nm * m3.x);  atomicAdd(d + 13, nm * m3.y);
  atomicAdd(d + 14, nm * m3.z);  atomicAdd(d + 15, nm * m3.w);
}

__global__ void relu_kernel(float* __restrict__ p, int n) {
  int i = blockIdx.x * blockDim.x + threadIdx.x;
  if (i < n) p[i] = fmaxf(p[i], 0.0f);
}

// ------------------------------------------------------------------ driver
extern "C" void kernel_launch(void* const* d_in, const int* in_sizes, int n_in,
                              void* d_out, int out_size, void* d_ws, size_t ws_size,
                              hipStream_t stream) {
  const float*     x  = (const float*)d_in[0];
  const long long* ei = (const long long*)d_in[1];   // [2, E] int64
  const float*     W1 = (const float*)d_in[2];
  const float*     V1 = (const float*)d_in[3];
  const float*     b1 = (const float*)d_in[4];
  const float*     W2 = (const float*)d_in[5];
  const float*     V2 = (const float*)d_in[6];
  const float*     b2 = (const float*)d_in[7];
  float*           out = (float*)d_out;

  const int N = in_sizes[0] / 8;   // 100000
  const int E = in_sizes[1] / 2;   // 3200000
  const long long* row = ei;       // source
  const long long* col = ei + E;   // target

  // Workspace layout (fp32): dinv[N] | norm[E] | xw[N*16] | acc[N*16]  (~26MB)
  float* ws   = (float*)d_ws;
  float* dinv = ws;
  float* norm = dinv + N;
  float* xw   = norm + E;
  float* acc  = xw + (size_t)N * 16;

  const int TB = 256;
  auto cdiv = [](int a, int b) { return (a + b - 1) / b; };

  // --- GCN normalization: deg over col, norm_e = dinv[row]*dinv[col] ---
  zero_f32_kernel<<<cdiv(N, TB), TB, 0, stream>>>(dinv, N);
  deg_accum_kernel<<<cdiv(E, TB), TB, 0, stream>>>(col, dinv, E);
  dinv_kernel<<<cdiv(N, TB), TB, 0, stream>>>(dinv, N);
  norm_kernel<<<cdiv(E, TB), TB, 0, stream>>>(row, col, dinv, norm, E);

  const int nwaves     = cdiv(N, 16);
  const int gemmBlocks = cdiv(nwaves * 32, TB);

  // --- Layer 1: xw = x@W1 ; acc = x@V1 + b1 ; acc += scatter(norm*xw) ; relu
  arma_dense_kernel<8><<<gemmBlocks, TB, 0, stream>>>(x, W1, V1, b1, xw, acc, N);
  scatter_kernel<<<cdiv(E, TB), TB, 0, stream>>>(row, col, norm, xw, acc, E);
  relu_kernel<<<cdiv(N * 16, TB), TB, 0, stream>>>(acc, N * 16);   // acc == h1

  // --- Layer 2: xw = h1@W2 ; out = h1@V2 + b2 ; out += scatter(norm*xw) ; relu
  arma_dense_kernel<16><<<gemmBlocks, TB, 0, stream>>>(acc, W2, V2, b2, xw, out, N);
  scatter_kernel<<<cdiv(E, TB), TB, 0, stream>>>(row, col, norm, xw, out, E);
  relu_kernel<<<cdiv(N * 16, TB), TB, 0, stream>>>(out, N * 16);
}